// TTM_2980707303683
// MI455X (gfx1250) — compile-verified
//
#include <hip/hip_runtime.h>
#include <math.h>

// ---------------- problem constants ----------------
#define A_DIM 16
#define B_DIM 128
#define T_DIM 512
#define M_DIM 2048
#define H_DIM 8192
#define TKN   3          // TK blocks
#define AVGF  32
#define LN_EPS 1e-5f

// ---------------- persistent-kernel config ----------------
#define NWG 32
#define BS  256
#define NWAVES (NWG * (BS / 32))   // 256 waves
#define NT (NWG * BS)              // 8192 threads

typedef __attribute__((ext_vector_type(2))) float v2f;
typedef __attribute__((ext_vector_type(8))) float v8f;

// CDNA5 async global->LDS copy path (ASYNCcnt-tracked), with safe fallback.
#if defined(__has_builtin)
#if __has_builtin(__builtin_amdgcn_global_load_async_to_lds_b32) && \
    __has_builtin(__builtin_amdgcn_s_wait_asynccnt)
#define HAVE_ASYNC_LDS 1
#endif
#endif

#ifdef HAVE_ASYNC_LDS
typedef __attribute__((address_space(1))) int g_int;   // global
typedef __attribute__((address_space(3))) int l_int;   // LDS
#endif

// ============================================================
// Kernel 1: xcf[f, m] = mean over 16 time steps of x[a, b, f*16+s]
// m = b*16 + a  (from transpose(2,1,0) then reshape)
// ============================================================
__global__ void k_reduce(const float* __restrict__ x, float* __restrict__ xcf) {
    int idx = blockIdx.x * blockDim.x + threadIdx.x;    // 0 .. 32*2048-1
    if (idx >= AVGF * M_DIM) return;
    int f = idx >> 11;
    int m = idx & (M_DIM - 1);
    int aa = m & 15;          // a index
    int bb = m >> 4;          // b index
    const float* p = x + (size_t)aa * (B_DIM * T_DIM) + (size_t)bb * T_DIM + f * 16;
    float s = 0.f;
#pragma unroll
    for (int t = 0; t < 16; ++t) s += p[t];
    xcf[idx] = s * (1.0f / 16.0f);
}

// ============================================================
// Kernel 2: s0 = xcf @ weight^T + sinusoidal bias via
// V_WMMA_F32_16X16X4_F32 (fp32 WMMA, no precision loss).
// One wave per 16x16 output tile.  ISA 32-bit layouts:
// lanes 0-15 hold K0/K1, lanes 16-31 hold K2/K3 for both A and B.
// ============================================================
__global__ void __launch_bounds__(256) k_s0_wmma(const float* __restrict__ xcf,
                                                 const float* __restrict__ weight,
                                                 float* __restrict__ s0) {
    int wid  = blockIdx.x * (blockDim.x >> 5) + (threadIdx.x >> 5);  // 0..255
    int lane = threadIdx.x & 31;
    int ft = wid >> 7;        // 0..1  (token tile)
    int rt = wid & 127;       // 0..127 (output-col tile)
    int half = lane >> 4;     // 0: K0/K1, 1: K2/K3
    int l15  = lane & 15;

    // Accumulator seeded with positional bias (D layout: VGPR r -> M=r+half*8, N=l15)
    v8f acc;
#pragma unroll
    for (int r = 0; r < 8; ++r) {
        int fI  = ft * 16 + r + half * 8;
        int col = rt * 16 + l15;
        int me  = col & ~1;
        float ang = (float)fI * powf(10000.0f, -2.0f * (float)me / (float)M_DIM);
        acc[r] = (col & 1) ? cosf(ang) : sinf(ang);
    }

    const float* arow = xcf    + (size_t)(ft * 16 + l15) * M_DIM + half * 2;
    const float* brow = weight + (size_t)(rt * 16 + l15) * M_DIM + half * 2;

    for (int k0 = 0; k0 < M_DIM; k0 += 4) {
        v2f a, b;
        a.x = arow[k0]; a.y = arow[k0 + 1];
        b.x = brow[k0]; b.y = brow[k0 + 1];
        acc = __builtin_amdgcn_wmma_f32_16x16x4_f32(false, a, false, b,
                                                    (short)0, acc, false, false);
    }

#pragma unroll
    for (int r = 0; r < 8; ++r) {
        int fI  = ft * 16 + r + half * 8;
        int col = rt * 16 + l15;
        s0[(size_t)fI * M_DIM + col] = acc[r];
    }
}

// ============================================================
// Kernel 3: persistent sequential chain (32 tokens x 3 blocks).
// softmax over a singleton axis == 1 => q/k never affect the output,
// so only the Wv / Wo / fc1 / fc2 matvecs remain.
// ============================================================
__device__ __forceinline__ float wave_sum(float v) {
#pragma unroll
    for (int o = 16; o > 0; o >>= 1) v += __shfl_xor(v, o, 32);
    return v;
}

__device__ __forceinline__ float block_sum(float v, volatile float* red) {
    float w = wave_sum(v);
    if ((threadIdx.x & 31) == 0) red[threadIdx.x >> 5] = w;
    __syncthreads();
    float s = 0.f;
#pragma unroll
    for (int i = 0; i < BS / 32; ++i) s += red[i];   // same order in every WG
    __syncthreads();
    return s;
}

__device__ __forceinline__ void gbar(unsigned* cnt, unsigned* gen) {
    __threadfence();
    __syncthreads();
    if (threadIdx.x == 0) {
        unsigned g = __atomic_load_n(gen, __ATOMIC_RELAXED);
        if (__atomic_fetch_add(cnt, 1u, __ATOMIC_ACQ_REL) == (unsigned)(NWG - 1)) {
            __atomic_store_n(cnt, 0u, __ATOMIC_RELAXED);
            __atomic_fetch_add(gen, 1u, __ATOMIC_ACQ_REL);
        } else {
            while (__atomic_load_n(gen, __ATOMIC_ACQUIRE) == g)
                __builtin_amdgcn_s_sleep(2);
        }
    }
    __syncthreads();
    __threadfence();
}

// Stage a plain global vector into LDS: async DMA path on CDNA5 if available.
__device__ __forceinline__ void stage_lds(float* lds_dst, const float* gsrc, int n) {
#ifdef HAVE_ASYNC_LDS
    l_int* ldst = (l_int*)lds_dst;
    for (int i = threadIdx.x; i < n; i += BS)
        __builtin_amdgcn_global_load_async_to_lds_b32(
            (g_int*)(gsrc + i), ldst + i, 0, 0);
    __builtin_amdgcn_s_wait_asynccnt(0);
#else
    for (int i = threadIdx.x; i < n; i += BS) lds_dst[i] = gsrc[i];
#endif
    __syncthreads();
}

__global__ void __launch_bounds__(BS) k_chain(
    const float* __restrict__ Wv,  const float* __restrict__ Wo,
    const float* __restrict__ g1,  const float* __restrict__ b1,
    const float* __restrict__ g2,  const float* __restrict__ b2,
    const float* __restrict__ fc1w, const float* __restrict__ fc1b,
    const float* __restrict__ fc2w, const float* __restrict__ fc2b,
    const float* __restrict__ s0,
    float* svec, float* t1, float* hbuf, float* vcum,
    float* dout, unsigned* cnt, unsigned* gen)
{
    __shared__ float red[BS / 32];
    __shared__ float op[M_DIM];       // 8 KB operand buffer (xn / vcum / s2)
    __shared__ float oph[H_DIM];      // 32 KB operand buffer (h)
    const int lane = threadIdx.x & 31;
    const int wid  = blockIdx.x * (BS / 32) + (threadIdx.x >> 5);
    const int gtid = blockIdx.x * BS + threadIdx.x;
    const float4* op4  = (const float4*)op;
    const float4* oph4 = (const float4*)oph;

    for (int f = 0; f < AVGF; ++f) {
        // token head: svec = s0[f]; zero vcum on first token
        for (int i = gtid; i < M_DIM; i += NT) svec[i] = s0[f * M_DIM + i];
        if (f == 0)
            for (int i = gtid; i < TKN * M_DIM; i += NT) vcum[i] = 0.f;
        gbar(cnt, gen);

        for (int a = 0; a < TKN; ++a) {
            float* vc = vcum + a * M_DIM;

            // LN1 stats over svec (redundant per WG, identical order => deterministic)
            float ps = 0.f;
            for (int i = threadIdx.x; i < M_DIM; i += BS) ps += svec[i];
            float mu1 = block_sum(ps, red) * (1.f / M_DIM);
            float pv = 0.f;
            for (int i = threadIdx.x; i < M_DIM; i += BS) {
                float d = svec[i] - mu1; pv += d * d;
            }
            float is1 = 1.f / sqrtf(block_sum(pv, red) * (1.f / M_DIM) + LN_EPS);

            // stage xn = LN1(svec) into LDS once per WG
            for (int i = threadIdx.x; i < M_DIM; i += BS)
                op[i] = (svec[i] - mu1) * is1 * g1[i] + b1[i];
            __syncthreads();

            // phase 2: vcum[a] += Wv[a] @ xn
            const float* Wva = Wv + (size_t)a * M_DIM * M_DIM;
            for (int row = wid; row < M_DIM; row += NWAVES) {
                const float4* wr4 = (const float4*)(Wva + (size_t)row * M_DIM);
                float acc = 0.f;
                for (int m4 = lane; m4 < M_DIM / 4; m4 += 32) {
                    float4 w = wr4[m4], o = op4[m4];
                    acc += w.x * o.x + w.y * o.y + w.z * o.z + w.w * o.w;
                }
                acc = wave_sum(acc);
                if (lane == 0) vc[row] += acc;
            }
            gbar(cnt, gen);

            // phase 3: t1 = Wo[a] @ vcum[a] + svec    (vcum staged via async LDS copy)
            stage_lds(op, vc, M_DIM);
            const float* Woa = Wo + (size_t)a * M_DIM * M_DIM;
            for (int row = wid; row < M_DIM; row += NWAVES) {
                const float4* wr4 = (const float4*)(Woa + (size_t)row * M_DIM);
                float acc = 0.f;
                for (int m4 = lane; m4 < M_DIM / 4; m4 += 32) {
                    float4 w = wr4[m4], o = op4[m4];
                    acc += w.x * o.x + w.y * o.y + w.z * o.z + w.w * o.w;
                }
                acc = wave_sum(acc);
                if (lane == 0) t1[row] = acc + svec[row];
            }
            gbar(cnt, gen);

            // LN2 stats over t1
            ps = 0.f;
            for (int i = threadIdx.x; i < M_DIM; i += BS) ps += t1[i];
            float mu2 = block_sum(ps, red) * (1.f / M_DIM);
            pv = 0.f;
            for (int i = threadIdx.x; i < M_DIM; i += BS) {
                float d = t1[i] - mu2; pv += d * d;
            }
            float is2 = 1.f / sqrtf(block_sum(pv, red) * (1.f / M_DIM) + LN_EPS);

            // stage s2 = LN2(t1) + t1 into LDS once per WG
            for (int i = threadIdx.x; i < M_DIM; i += BS) {
                float tm = t1[i];
                op[i] = (tm - mu2) * is2 * g2[i] + b2[i] + tm;
            }
            __syncthreads();

            // phase 4: hbuf = gelu(fc1w @ s2 + fc1b), exact gelu
            for (int row = wid; row < H_DIM; row += NWAVES) {
                const float4* wr4 = (const float4*)(fc1w + (size_t)row * M_DIM);
                if (row + NWAVES < H_DIM)
                    __builtin_prefetch(fc1w + (size_t)(row + NWAVES) * M_DIM + lane * 4, 0, 0);
                float acc = 0.f;
                for (int m4 = lane; m4 < M_DIM / 4; m4 += 32) {
                    float4 w = wr4[m4], o = op4[m4];
                    acc += w.x * o.x + w.y * o.y + w.z * o.z + w.w * o.w;
                }
                acc = wave_sum(acc);
                if (lane == 0) {
                    float z = acc + fc1b[row];
                    hbuf[row] = 0.5f * z * (1.f + erff(z * 0.70710678118654752f));
                }
            }
            gbar(cnt, gen);

            // phase 5: s = fc2w @ h + fc2b   (h staged via async LDS copy)
            stage_lds(oph, hbuf, H_DIM);
            for (int row = wid; row < M_DIM; row += NWAVES) {
                const float4* wr4 = (const float4*)(fc2w + (size_t)row * H_DIM);
                float acc = 0.f;
                for (int m4 = lane; m4 < H_DIM / 4; m4 += 32) {
                    float4 w = wr4[m4], o = oph4[m4];
                    acc += w.x * o.x + w.y * o.y + w.z * o.z + w.w * o.w;
                }
                acc = wave_sum(acc);
                if (lane == 0) {
                    float val = acc + fc2b[row];
                    if (a == TKN - 1) dout[f * M_DIM + row] = val;
                    else              svec[row] = val;
                }
            }
            gbar(cnt, gen);
        }
    }
}

// ============================================================
// launcher
// ============================================================
extern "C" void kernel_launch(void* const* d_in, const int* in_sizes, int n_in,
                              void* d_out, int out_size, void* d_ws, size_t ws_size,
                              hipStream_t stream) {
    (void)in_sizes; (void)n_in; (void)out_size; (void)ws_size;
    const float* x      = (const float*)d_in[0];
    const float* weight = (const float*)d_in[1];
    // d_in[2] = Wq, d_in[3] = Wk: provably unused (softmax over singleton == 1)
    const float* Wv     = (const float*)d_in[4];
    const float* Wo     = (const float*)d_in[5];
    const float* g1     = (const float*)d_in[6];
    const float* b1     = (const float*)d_in[7];
    const float* g2     = (const float*)d_in[8];
    const float* b2     = (const float*)d_in[9];
    const float* fc1w   = (const float*)d_in[10];
    const float* fc1b   = (const float*)d_in[11];
    const float* fc2w   = (const float*)d_in[12];
    const float* fc2b   = (const float*)d_in[13];
    float* out = (float*)d_out;

    float* ws   = (float*)d_ws;
    float* XCF  = ws;                 // 32*2048
    float* S0B  = ws + 65536;         // 32*2048
    float* VCUM = ws + 131072;        // 3*2048
    float* SVEC = ws + 137216;        // 2048
    float* T1   = ws + 139264;        // 2048
    float* HB   = ws + 141312;        // 8192
    unsigned* CNT = (unsigned*)(ws + 149504);
    unsigned* GEN = CNT + 1;

    (void)hipMemsetAsync(CNT, 0, 2 * sizeof(unsigned), stream);  // graph-capture safe
    k_reduce <<<dim3(256), dim3(256), 0, stream>>>(x, XCF);
    k_s0_wmma<<<dim3(32),  dim3(256), 0, stream>>>(XCF, weight, S0B);
    k_chain  <<<dim3(NWG), dim3(BS),  0, stream>>>(Wv, Wo, g1, b1, g2, b2,
                                                   fc1w, fc1b, fc2w, fc2b,
                                                   S0B, SVEC, T1, HB, VCUM,
                                                   out, CNT, GEN);
}